// SimpleAttention_37529424233348
// MI455X (gfx1250) — compile-verified
//
#include <hip/hip_runtime.h>

// Reference shapes: query [B,D] f32, values [B,T,D] f32.
constexpr int B = 32;
constexpr int T = 8192;
constexpr int D = 256;
constexpr int TC = 512;                 // t-rows per workgroup chunk
constexpr int NCHUNK = T / TC;          // 16 chunks per batch
constexpr int ROWS_PER_WAVE = TC / 8;   // 64 rows per wave (4 tiles of 16)

typedef __attribute__((ext_vector_type(2))) float v2f;
typedef __attribute__((ext_vector_type(4))) float v4f;
typedef __attribute__((ext_vector_type(8))) float v8f;

// ---------------------------------------------------------------------------
// Fused pass: per 16-row tile, (1) score via V_WMMA_F32_16X16X4_F32,
// (2) online-softmax update of running (mx,sum), (3) rank-16 context update
// with VALU FMAs re-reading the cache-hot tile. One HBM pass over values.
//
// WMMA f32 16x16x4 layouts (CDNA5 ISA):
//   A 16x4: lane<16 -> (M=lane, K=0,1); lane>=16 -> (M=lane-16, K=2,3)
//   B 4x16: lane<16 -> rows K=0,1;     lane>=16 -> rows K=2,3   (N=lane&15)
//   C:      VGPR r, lanes 0-15 = (M=r,N=lane); lanes 16-31 = (M=8+r,N=lane-16)
// With B broadcast over N, C columns are replicated: lanes<16 all hold
// s[0..7], lanes>=16 all hold s[8..15]; one shfl_xor(16) completes the set.
// ---------------------------------------------------------------------------
__global__ __launch_bounds__(256) void fused_attn_kernel(
    const float* __restrict__ q, const float* __restrict__ V,
    float* __restrict__ score,                     // [B*T] raw scores (ws)
    float* __restrict__ cmx, float* __restrict__ csum,  // [B*NCHUNK]
    float* __restrict__ cctx)                      // [B*NCHUNK*D]
{
  __shared__ float smx[8], ssum[8];
  __shared__ float sctx[8][D];

  const int tid   = threadIdx.x;
  const int lane  = tid & 31;
  const int wv    = tid >> 5;
  const int chunk = blockIdx.x;                    // [0, B*NCHUNK)
  const int b     = chunk / NCHUNK;
  const int c0    = (chunk - b * NCHUNK) * TC;

  const int m  = lane & 15;
  const int kh = (lane >> 4) << 1;                 // 0 or 2

  const float* qk = q + (size_t)b * D + kh;
  const float* Vb = V + (size_t)b * T * D;

  float mx  = -__builtin_inff();
  float sum = 0.f;
  float ctx0[8];                                   // lane owns d = lane*8..+7
#pragma unroll
  for (int j = 0; j < 8; ++j) ctx0[j] = 0.f;

  const int waveBase = c0 + wv * ROWS_PER_WAVE;

  for (int tt = 0; tt < ROWS_PER_WAVE; tt += 16) {
    const int t0 = waveBase + tt;

    // ---- score tile: WMMA GEMV, K-loop over D ----
    const float* vrow = Vb + (size_t)(t0 + m) * D + kh;
    v8f c = {};
#pragma unroll 8
    for (int k = 0; k < D; k += 4) {
      v2f a  = *(const v2f*)(vrow + k);            // V[b][t0+m][k+kh..+1]
      v2f bb = *(const v2f*)(qk + k);              // q broadcast over N
      c = __builtin_amdgcn_wmma_f32_16x16x4_f32(false, a, false, bb,
                                                (short)0, c, false, false);
    }

    // stash raw scores for the weights pass (column N=0: lanes 0 and 16)
    if (m == 0) {
      float* sp = score + (size_t)b * T + t0 + (lane >> 4) * 8;
#pragma unroll
      for (int r = 0; r < 8; ++r) sp[r] = c[r];
    }

    // every lane assembles all 16 tile scores (swap wave halves)
    float s16[16];
#pragma unroll
    for (int r = 0; r < 8; ++r) {
      const float mine  = c[r];
      const float other = __shfl_xor(mine, 16, 32);
      if (lane < 16) { s16[r] = mine;  s16[r + 8] = other; }
      else           { s16[r + 8] = mine;  s16[r] = other; }
    }

    // ---- online softmax update ----
    float tmax = s16[0];
#pragma unroll
    for (int r = 1; r < 16; ++r) tmax = fmaxf(tmax, s16[r]);
    const float nmx  = fmaxf(mx, tmax);
    const float scal = __expf(mx - nmx);
    float e[16];
    float esum = 0.f;
#pragma unroll
    for (int r = 0; r < 16; ++r) { e[r] = __expf(s16[r] - nmx); esum += e[r]; }
    sum = sum * scal + esum;
    mx  = nmx;
#pragma unroll
    for (int j = 0; j < 8; ++j) ctx0[j] *= scal;

    // ---- rank-16 context update: rows just streamed -> WGP$/L2 hot.
    // Lane owns 32B of each row: full-wave 1KB coalesced loads.
    const float* vr = Vb + (size_t)t0 * D + lane * 8;
#pragma unroll 4
    for (int r = 0; r < 16; ++r) {
      const v4f lo = *(const v4f*)(vr + (size_t)r * D);
      const v4f hi = *(const v4f*)(vr + (size_t)r * D + 4);
      ctx0[0] += e[r] * lo.x; ctx0[1] += e[r] * lo.y;
      ctx0[2] += e[r] * lo.z; ctx0[3] += e[r] * lo.w;
      ctx0[4] += e[r] * hi.x; ctx0[5] += e[r] * hi.y;
      ctx0[6] += e[r] * hi.z; ctx0[7] += e[r] * hi.w;
    }
  }

  // ---- combine the 8 wave partials in LDS -> one partial per WG ----
  if (lane == 0) { smx[wv] = mx; ssum[wv] = sum; }
#pragma unroll
  for (int j = 0; j < 8; ++j) sctx[wv][lane * 8 + j] = ctx0[j];
  __syncthreads();

  float mx8 = smx[0];
#pragma unroll
  for (int i = 1; i < 8; ++i) mx8 = fmaxf(mx8, smx[i]);
  float sum8 = 0.f, cacc = 0.f;
#pragma unroll
  for (int i = 0; i < 8; ++i) {
    const float sc = __expf(smx[i] - mx8);
    sum8 += sc * ssum[i];
    cacc += sc * sctx[i][tid];
  }
  cctx[(size_t)chunk * D + tid] = cacc;
  if (tid == 0) { cmx[chunk] = mx8; csum[chunk] = sum8; }
}

// ---------------------------------------------------------------------------
// Per-batch finalize: merge NCHUNK chunk partials -> global (mx,sum), write
// context, then stream weights = exp(score - mx)/sum. ~3 MB total traffic.
// ---------------------------------------------------------------------------
__global__ __launch_bounds__(256) void finalize_kernel(
    const float* __restrict__ score, const float* __restrict__ cmx,
    const float* __restrict__ csum, const float* __restrict__ cctx,
    float* __restrict__ context, float* __restrict__ weights)
{
  const int b   = blockIdx.x;
  const int tid = threadIdx.x;

  float mxg = -__builtin_inff();
#pragma unroll
  for (int i = 0; i < NCHUNK; ++i)
    mxg = fmaxf(mxg, cmx[b * NCHUNK + i]);

  float sumg = 0.f, acc = 0.f;
#pragma unroll
  for (int i = 0; i < NCHUNK; ++i) {
    const float sc = __expf(cmx[b * NCHUNK + i] - mxg);
    sumg += sc * csum[b * NCHUNK + i];
    acc  += sc * cctx[(size_t)(b * NCHUNK + i) * D + tid];
  }
  context[(size_t)b * D + tid] = acc;

  const float inv = 1.f / sumg;
  const float* s = score + (size_t)b * T;
  float* w = weights + (size_t)b * T;
  for (int t = tid; t < T; t += 256) w[t] = __expf(s[t] - mxg) * inv;
}

extern "C" void kernel_launch(void* const* d_in, const int* in_sizes, int n_in,
                              void* d_out, int out_size, void* d_ws, size_t ws_size,
                              hipStream_t stream) {
  const float* q = (const float*)d_in[0];             // [B, D]
  const float* V = (const float*)d_in[1];             // [B, T, D]
  float* context = (float*)d_out;                     // [B, D]   output 0
  float* weights = (float*)d_out + (size_t)B * D;     // [B, T]   output 1

  // workspace layout (floats): scores | cmx | csum | cctx
  float* wsScore = (float*)d_ws;                            // B*T      (1 MB)
  float* wsMx    = wsScore + (size_t)B * T;                 // B*NCHUNK
  float* wsSum   = wsMx + (size_t)B * NCHUNK;               // B*NCHUNK
  float* wsCtx   = wsSum + (size_t)B * NCHUNK;              // B*NCHUNK*D (512 KB)

  fused_attn_kernel<<<B * NCHUNK, 256, 0, stream>>>(q, V, wsScore, wsMx, wsSum, wsCtx);
  finalize_kernel<<<B, 256, 0, stream>>>(wsScore, wsMx, wsSum, wsCtx, context, weights);
}